// Graph_conv_layer_45449343926933
// MI455X (gfx1250) — compile-verified
//
#include <hip/hip_runtime.h>

typedef __attribute__((ext_vector_type(2))) float v2f;
typedef __attribute__((ext_vector_type(8))) float v8f;

#define N_NODES 100000
#define N_EDGES 1600000
#define IN_DIM  256
#define OUT_DIM 128
#define M_TILES (N_NODES / 16)      // 6250, exact
#define GEMM_BLOCKS 625             // 6250 / 625 = 10 M-tiles per block

// -----------------------------------------------------------------------------
// Kernel 1: support = X @ W with V_WMMA_F32_16X16X4_F32 (exact fp32).
// Persistent blocks: 625 blocks x 8 wave32. Wave w owns N-tile [16w,16w+16)
// and keeps its whole 256x16 panel of W in registers (v2f bfrag[64] = 128
// VGPRs/lane), loaded once. Then each M-tile costs only 64 A-loads (b64) +
// 64 WMMAs: 1 VMEM per WMMA in steady state.
//
// Fragment layouts (CDNA5 ISA 7.12.2):
//   A 16x4  (2 VGPRs): lane L -> M = L&15, K = v + (L>=16 ? 2 : 0)
//   B 4x16  (2 VGPRs): lane L -> N = L&15, K = v + (L>=16 ? 2 : 0)
//   C 16x16 (8 VGPRs): VGPR r, lane L -> M = r + (L>=16 ? 8 : 0), N = L&15
// -----------------------------------------------------------------------------
__global__ __launch_bounds__(256) void gemm_wmma_f32(const float* __restrict__ X,
                                                     const float* __restrict__ W,
                                                     float* __restrict__ S) {
  const int lane = threadIdx.x & 31;
  const int wave = threadIdx.x >> 5;        // 0..7 -> N tile index
  const int n0   = wave << 4;
  const int m    = lane & 15;               // doubles as this lane's N index for B/C
  const int koff = (lane >> 4) << 1;        // 0 for lanes 0-15, 2 for lanes 16-31
  const int mbase = (lane >> 4) << 3;       // 0 or 8 (C-fragment row base)

  // ---- Load this wave's B panel (W[:, n0:n0+16]) into registers, once. ----
  const float* __restrict__ wcol = W + (n0 + m);
  v2f bfrag[64];
#pragma unroll
  for (int kk = 0; kk < 64; ++kk) {
    const int k = (kk << 2) + koff;
    bfrag[kk].x = wcol[(size_t)k * OUT_DIM];
    bfrag[kk].y = wcol[(size_t)(k + 1) * OUT_DIM];
  }

  // ---- Grid-stride over M tiles: A loads are the only per-tile VMEM. ----
  for (int mt = blockIdx.x; mt < M_TILES; mt += gridDim.x) {
    const int row0 = mt << 4;
    const float* __restrict__ xrow = X + (size_t)(row0 + m) * IN_DIM + koff;

    v8f c = {0.f, 0.f, 0.f, 0.f, 0.f, 0.f, 0.f, 0.f};
#pragma unroll
    for (int kk = 0; kk < 64; ++kk) {
      const v2f a = *(const v2f*)(xrow + (kk << 2));   // K-contiguous pair: one b64
      c = __builtin_amdgcn_wmma_f32_16x16x4_f32(
          /*neg_a=*/false, a, /*neg_b=*/false, bfrag[kk],
          /*c_mod=*/(short)0, c, /*reuse_a=*/false, /*reuse_b=*/false);
    }

    float* __restrict__ srow = S + (size_t)row0 * OUT_DIM + n0 + m;
#pragma unroll
    for (int r = 0; r < 8; ++r) {
      srow[(size_t)(mbase + r) * OUT_DIM] = c[r];
    }
  }
}

// -----------------------------------------------------------------------------
// Kernel 2: out[n][d] = bias[d]  (also clears the 0xAA poison)
// -----------------------------------------------------------------------------
__global__ __launch_bounds__(256) void init_bias(float* __restrict__ out,
                                                 const float* __restrict__ bias,
                                                 long long n) {
  const long long i = (long long)blockIdx.x * blockDim.x + threadIdx.x;
  if (i < n) out[i] = bias[(int)(i & (OUT_DIM - 1))];
}

// -----------------------------------------------------------------------------
// Kernel 3: edge-parallel SpMM. 128 consecutive threads own one edge (edge
// index is wave-uniform); thread d does out[r][d] += v * S[c][d] with a
// relaxed agent-scope fp32 atomic (non-returning GLOBAL_ATOMIC_ADD_F32).
// S (51.2 MB) is resident in the 192 MB L2, so gather + atomics stay on-chip.
// -----------------------------------------------------------------------------
__global__ __launch_bounds__(256) void spmm_edge_atomic(const int* __restrict__ rows,
                                                        const int* __restrict__ cols,
                                                        const float* __restrict__ vals,
                                                        const float* __restrict__ S,
                                                        float* __restrict__ out,
                                                        int nEdges) {
  const long long gid = (long long)blockIdx.x * blockDim.x + threadIdx.x;
  const int e = (int)(gid >> 7);
  const int d = (int)(gid & 127);
  if (e >= nEdges) return;

  const int   r  = rows[e];
  const int   cc = cols[e];
  const float v  = vals[e];
  const float contrib = S[(size_t)cc * OUT_DIM + d] * v;

  __hip_atomic_fetch_add(&out[(size_t)r * OUT_DIM + d], contrib,
                         __ATOMIC_RELAXED, __HIP_MEMORY_SCOPE_AGENT);
}

// -----------------------------------------------------------------------------
extern "C" void kernel_launch(void* const* d_in, const int* in_sizes, int n_in,
                              void* d_out, int out_size, void* d_ws, size_t ws_size,
                              hipStream_t stream) {
  const float* X    = (const float*)d_in[0];   // [N_NODES, IN_DIM]
  const int*   er   = (const int*)  d_in[1];   // [N_EDGES]
  const int*   ec   = (const int*)  d_in[2];   // [N_EDGES]
  const float* ev   = (const float*)d_in[3];   // [N_EDGES]
  const float* W    = (const float*)d_in[4];   // [IN_DIM, OUT_DIM]
  const float* bias = (const float*)d_in[5];   // [OUT_DIM]

  float* out = (float*)d_out;                  // [N_NODES, OUT_DIM]
  float* S   = (float*)d_ws;                   // support scratch, 51.2 MB

  // 1) support = X @ W   (WMMA fp32, persistent blocks, B panel in registers)
  gemm_wmma_f32<<<GEMM_BLOCKS, 256, 0, stream>>>(X, W, S);

  // 2) out = bias
  const long long nOut = (long long)N_NODES * OUT_DIM;
  init_bias<<<(int)((nOut + 255) / 256), 256, 0, stream>>>(out, bias, nOut);

  // 3) out[r] += v * support[c]  over all edges
  const long long nWork = (long long)N_EDGES * OUT_DIM;   // 128 threads per edge
  spmm_edge_atomic<<<(int)(nWork / 256), 256, 0, stream>>>(er, ec, ev, S, out, N_EDGES);
}